// MyNodeGNN_80960133529605
// MI455X (gfx1250) — compile-verified
//
#include <hip/hip_runtime.h>
#include <hip/hip_bf16.h>

#define GNN_N 100000
#define GNN_E 1600000
#define HID 32
#define BN_EPS 1e-5f

typedef __attribute__((ext_vector_type(2))) float v2f;
typedef __attribute__((ext_vector_type(8))) float v8f;

// ---------------------------------------------------------------------------
// segment_sum scatter kernels (atomic fp32 adds)
// ---------------------------------------------------------------------------
__global__ void gin_scatter_c1(const float* __restrict__ x,
                               const int* __restrict__ src,
                               const int* __restrict__ dst,
                               float* __restrict__ agg) {
  int e = blockIdx.x * blockDim.x + threadIdx.x;
  if (e >= GNN_E) return;
  atomicAdd(&agg[dst[e]], x[src[e]]);
}

__global__ void gin_scatter_c32(const float* __restrict__ h,
                                const int* __restrict__ src,
                                const int* __restrict__ dst,
                                float* __restrict__ agg) {
  int idx = blockIdx.x * blockDim.x + threadIdx.x;  // GNN_E * 8 threads
  if (idx >= GNN_E * 8) return;
  int e  = idx >> 3;
  int c0 = (idx & 7) << 2;
  int s = src[e], d = dst[e];
  const float4 v = *reinterpret_cast<const float4*>(h + (size_t)s * HID + c0);
  float* o = agg + (size_t)d * HID + c0;
  atomicAdd(o + 0, v.x);
  atomicAdd(o + 1, v.y);
  atomicAdd(o + 2, v.z);
  atomicAdd(o + 3, v.w);
}

// ---------------------------------------------------------------------------
// layer-0 first linear: in_ch = 1, z[i][c] = (x[i]+agg[i])*W1[c] + b1[c]
// ---------------------------------------------------------------------------
__global__ void gin_lin0(const float* __restrict__ x, const float* __restrict__ agg,
                         const float* __restrict__ W1, const float* __restrict__ b1,
                         float* __restrict__ z) {
  int idx = blockIdx.x * blockDim.x + threadIdx.x;
  if (idx >= GNN_N * HID) return;
  int i = idx >> 5, c = idx & 31;
  float v = x[i] + agg[i];
  z[idx] = fmaf(v, W1[c], b1[c]);
}

// ---------------------------------------------------------------------------
// [N,32] x [32,32] + bias via V_WMMA_F32_16X16X4_F32 (fp32 matrix pipe).
// One wave computes a 16x32 output tile (two 16x16 accumulators), K chained
// in 8 steps of 4. ADD_AGG fuses the GIN (1+eps)*h + agg into the A-load.
// A frag (16x4 f32): lanes 0-15 hold K=k..k+1, lanes 16-31 hold K=k+2..k+3,
// row = lane&15.  B frag (4x16): VGPR j, lane half -> W[(k+hi*2+j)*32 + col].
// C/D: VGPR r -> row r (lanes 0-15) / r+8 (lanes 16-31), col = lane&15 + n0.
// ---------------------------------------------------------------------------
template <bool ADD_AGG>
__global__ void gin_linear_wmma(const float* __restrict__ A,
                                const float* __restrict__ AGG,
                                const float* __restrict__ W,
                                const float* __restrict__ bias,
                                float* __restrict__ out) {
  __shared__ float sW[HID * HID];
  __shared__ float sB[HID];
  int tid = threadIdx.x;
  for (int i = tid; i < HID * HID; i += blockDim.x) sW[i] = W[i];
  if (tid < HID) sB[tid] = bias[tid];
  __syncthreads();

  const int lane = tid & 31;
  const int wave = tid >> 5;
  const int tile = blockIdx.x * (blockDim.x >> 5) + wave;  // 16-row tile id
  const int row0 = tile * 16;
  if (row0 >= GNN_N) return;  // wave-uniform exit: EXEC stays all-1s for WMMA

  const int lh = lane & 15;  // row within tile (A), col within 16-col block (B/D)
  const int hi = lane >> 4;  // K sub-offset selector

  const float* arow = A + (size_t)(row0 + lh) * HID;
  const float* grow = ADD_AGG ? (AGG + (size_t)(row0 + lh) * HID) : nullptr;

  v8f acc0 = {};
  v8f acc1 = {};
#pragma unroll
  for (int k0 = 0; k0 < HID; k0 += 4) {
    const int k = k0 + hi * 2;
    v2f a;
    a.x = arow[k + 0];
    a.y = arow[k + 1];
    if (ADD_AGG) {
      a.x += grow[k + 0];
      a.y += grow[k + 1];
    }
    v2f b0, b1;
    b0.x = sW[(k + 0) * HID + 0 + lh];
    b0.y = sW[(k + 1) * HID + 0 + lh];
    b1.x = sW[(k + 0) * HID + 16 + lh];
    b1.y = sW[(k + 1) * HID + 16 + lh];
    acc0 = __builtin_amdgcn_wmma_f32_16x16x4_f32(false, a, false, b0, (short)0,
                                                 acc0, false, false);
    acc1 = __builtin_amdgcn_wmma_f32_16x16x4_f32(false, a, false, b1, (short)0,
                                                 acc1, false, false);
  }

#pragma unroll
  for (int r = 0; r < 8; ++r) {
    const int row = row0 + r + hi * 8;
    out[(size_t)row * HID + 0 + lh]  = acc0[r] + sB[0 + lh];
    out[(size_t)row * HID + 16 + lh] = acc1[r] + sB[16 + lh];
  }
}

// ---------------------------------------------------------------------------
// BatchNorm (training mode, biased variance) over the node dimension
// ---------------------------------------------------------------------------
__global__ void gin_bn_stats(const float* __restrict__ z, float* __restrict__ sums) {
  __shared__ float s_sum[HID];
  __shared__ float s_sq[HID];
  int tid = threadIdx.x;
  if (tid < HID) { s_sum[tid] = 0.f; s_sq[tid] = 0.f; }
  __syncthreads();
  const int c = tid & 31;  // grid stride is a multiple of 32 -> channel fixed
  float lsum = 0.f, lsq = 0.f;
  const int total = GNN_N * HID;
  for (int idx = blockIdx.x * blockDim.x + tid; idx < total;
       idx += gridDim.x * blockDim.x) {
    float v = z[idx];
    lsum += v;
    lsq = fmaf(v, v, lsq);
  }
  atomicAdd(&s_sum[c], lsum);
  atomicAdd(&s_sq[c], lsq);
  __syncthreads();
  if (tid < HID) {
    atomicAdd(&sums[tid], s_sum[tid]);
    atomicAdd(&sums[HID + tid], s_sq[tid]);
  }
}

__global__ void gin_bn_finalize(const float* __restrict__ sums,
                                const float* __restrict__ g,
                                const float* __restrict__ be,
                                float* __restrict__ scsh, float inv_n) {
  int c = threadIdx.x;
  if (c < HID) {
    float mean = sums[c] * inv_n;
    float var  = fmaf(-mean, mean, sums[HID + c] * inv_n);
    float sc   = g[c] * rsqrtf(var + BN_EPS);
    scsh[c]       = sc;
    scsh[HID + c] = fmaf(-mean, sc, be[c]);
  }
}

__global__ void gin_bn_apply_relu(float* __restrict__ z,
                                  const float* __restrict__ scsh) {
  int idx = blockIdx.x * blockDim.x + threadIdx.x;
  if (idx >= GNN_N * HID) return;
  int c = idx & 31;
  float v = fmaf(z[idx], scsh[c], scsh[HID + c]);
  z[idx] = fmaxf(v, 0.0f);
}

// ---------------------------------------------------------------------------
// JK-cat head: out = [x | xs0..xs3] @ Wout[129,2] + bout
// ---------------------------------------------------------------------------
__global__ void gin_head(const float* __restrict__ x, const float* __restrict__ xs,
                         const float* __restrict__ Wout,
                         const float* __restrict__ bout, float* __restrict__ out) {
  __shared__ float sW[129 * 2];
  __shared__ float sb[2];
  for (int i = threadIdx.x; i < 129 * 2; i += blockDim.x) sW[i] = Wout[i];
  if (threadIdx.x < 2) sb[threadIdx.x] = bout[threadIdx.x];
  __syncthreads();
  int i = blockIdx.x * blockDim.x + threadIdx.x;
  if (i >= GNN_N) return;
  float a0 = sb[0], a1 = sb[1];
  float xv = x[i];
  a0 = fmaf(xv, sW[0], a0);
  a1 = fmaf(xv, sW[1], a1);
#pragma unroll
  for (int l = 0; l < 4; ++l) {
    const float* row = xs + (size_t)l * GNN_N * HID + (size_t)i * HID;
    const float* w = sW + 2 * (1 + HID * l);
    for (int c = 0; c < HID; ++c) {
      float v = row[c];
      a0 = fmaf(v, w[2 * c + 0], a0);
      a1 = fmaf(v, w[2 * c + 1], a1);
    }
  }
  out[2 * i + 0] = a0;
  out[2 * i + 1] = a1;
}

// ---------------------------------------------------------------------------
// Launch
// ---------------------------------------------------------------------------
extern "C" void kernel_launch(void* const* d_in, const int* in_sizes, int n_in,
                              void* d_out, int out_size, void* d_ws, size_t ws_size,
                              hipStream_t stream) {
  (void)in_sizes; (void)n_in; (void)out_size; (void)ws_size;

  const float* x  = (const float*)d_in[0];
  const int*   ei = (const int*)d_in[1];
  const int* src = ei;            // edge_index[0]
  const int* dst = ei + GNN_E;    // edge_index[1]

  // params: 4 layers x {W1,b1,g1,be1,W2,b2,g3,be3}, then Wout, bout
  const float* Lp[4][8];
  int p = 2;
  for (int l = 0; l < 4; ++l)
    for (int j = 0; j < 8; ++j) Lp[l][j] = (const float*)d_in[p++];
  const float* Wout = (const float*)d_in[p++];
  const float* bout = (const float*)d_in[p++];

  float* ws = (float*)d_ws;
  const size_t slab = (size_t)GNN_N * HID;
  float* xs    = ws;             // 4 JK slabs [N,32] each
  float* z1    = ws + 4 * slab;  // MLP hidden
  float* agg   = ws + 5 * slab;  // segment_sum (layer0 uses first N floats)
  float* stats = ws + 6 * slab;  // 64 floats: sum[32], sumsq[32]
  float* scsh  = stats + 64;     // 64 floats: scale[32], shift[32]

  const float inv_n = 1.0f / (float)GNN_N;
  const int tiles = (GNN_N + 15) / 16;
  const int grid_wmma = (tiles + 3) / 4;  // 4 waves (128 threads) per block
  const int grid_elem = (GNN_N * HID + 255) / 256;

  auto bn = [&](float* z, const float* g, const float* be) {
    hipMemsetAsync(stats, 0, 64 * sizeof(float), stream);
    gin_bn_stats<<<1024, 256, 0, stream>>>(z, stats);
    gin_bn_finalize<<<1, 32, 0, stream>>>(stats, g, be, scsh, inv_n);
    gin_bn_apply_relu<<<grid_elem, 256, 0, stream>>>(z, scsh);
  };

  // ---- layer 0 (in_ch = 1) ----
  hipMemsetAsync(agg, 0, GNN_N * sizeof(float), stream);
  gin_scatter_c1<<<(GNN_E + 255) / 256, 256, 0, stream>>>(x, src, dst, agg);
  gin_lin0<<<grid_elem, 256, 0, stream>>>(x, agg, Lp[0][0], Lp[0][1], z1);
  bn(z1, Lp[0][2], Lp[0][3]);
  gin_linear_wmma<false><<<grid_wmma, 128, 0, stream>>>(z1, nullptr, Lp[0][4],
                                                        Lp[0][5], xs);
  bn(xs, Lp[0][6], Lp[0][7]);

  // ---- layers 1..3 (32 channels) ----
  for (int l = 1; l < 4; ++l) {
    float* h = xs + (size_t)(l - 1) * slab;
    float* o = xs + (size_t)l * slab;
    hipMemsetAsync(agg, 0, slab * sizeof(float), stream);
    gin_scatter_c32<<<(GNN_E * 8 + 255) / 256, 256, 0, stream>>>(h, src, dst, agg);
    gin_linear_wmma<true><<<grid_wmma, 128, 0, stream>>>(h, agg, Lp[l][0],
                                                         Lp[l][1], z1);
    bn(z1, Lp[l][2], Lp[l][3]);
    gin_linear_wmma<false><<<grid_wmma, 128, 0, stream>>>(z1, nullptr, Lp[l][4],
                                                          Lp[l][5], o);
    bn(o, Lp[l][6], Lp[l][7]);
  }

  // ---- head ----
  gin_head<<<(GNN_N + 255) / 256, 256, 0, stream>>>(x, xs, Wout, bout,
                                                    (float*)d_out);
}